// WindowAttention_850403524730
// MI455X (gfx1250) — compile-verified
//
#include <hip/hip_runtime.h>
#include <hip/hip_bf16.h>

// ---------------------------------------------------------------------------
// Fused 3D window attention for MI455X (gfx1250), f32 WMMA 16x16x4 throughout.
// One wave = one 16-row tile = 2 windows. 8 waves per block = 16 windows.
// ---------------------------------------------------------------------------

typedef __attribute__((ext_vector_type(2))) float v2f;
typedef __attribute__((ext_vector_type(8))) float v8f;
typedef __attribute__((ext_vector_type(4))) int v4i;

#define XS 68            // x / attn-out LDS row stride (floats), bank-skewed
#define QS 210           // qkv LDS row stride (floats), even, bank-skewed
#define PS 20            // P (probabilities) LDS row stride
#define SHARED_F (12288 + 4096 + 512 + 192 + 64)     // swizzled weights + bias
#define WAVE_F   (16*XS + 16*QS + 16*PS + 128)       // per-wave scratch = 4896
#define TOTAL_F  (SHARED_F + 8*WAVE_F)               // 56320 floats = 220 KB

#if defined(__has_builtin)
# if __has_builtin(__builtin_amdgcn_global_load_async_to_lds_b128)
#  define HAVE_ASYNC_LDS 1
# endif
#endif

__device__ __forceinline__ v8f splat8(float v) {
  v8f r = {v, v, v, v, v, v, v, v};
  return r;
}

__device__ __forceinline__ v8f wmma4(v2f a, v2f b, v8f c) {
#if defined(__has_builtin) && __has_builtin(__builtin_amdgcn_wmma_f32_16x16x4_f32)
  return __builtin_amdgcn_wmma_f32_16x16x4_f32(false, a, false, b, (short)0, c,
                                               false, false);
#else
  // Placeholder so the file still compiles if the builtin is missing; the
  // histogram (wmma==0) will reveal this immediately.
  c[0] += a[0] * b[0];
  return c;
#endif
}

__global__ __launch_bounds__(256, 1) void wattn_kernel(
    const float* __restrict__ x, const float* __restrict__ mask,
    const float* __restrict__ qkv_w, const float* __restrict__ qkv_b,
    const float* __restrict__ proj_w, const float* __restrict__ proj_b,
    const float* __restrict__ bias_table, const int* __restrict__ rel_idx,
    float* __restrict__ out) {
  extern __shared__ float smem[];
  float* wq  = smem;             // [12][16][32] float2  B-frags of qkv_w
  float* pw  = wq + 12288;       // [4][16][32] float2   B-frags of proj_w
  float* bia = pw + 4096;        // [8][8][8]            bias[h][q][k]
  float* qkb = bia + 512;        // [192]
  float* prb = qkb + 192;        // [64]
  float* wav = prb + 64;         // per-wave scratch

  const int tid  = threadIdx.x;
  const int lane = tid & 31;
  const int wv   = tid >> 5;
  const int mrow = lane & 15;    // M row (A/C) or N col (B/C), per layout
  const int hi   = lane >> 4;    // half-wave select

  // ---- cooperative: swizzle weights into exact WMMA B-fragment layout ----
  for (int i = tid; i < 12 * 16 * 32; i += 256) {
    int n = i >> 9, k4 = (i >> 5) & 15, ln = i & 31;
    int row = 4 * k4 + 2 * (ln >> 4), col = n * 16 + (ln & 15);
    wq[2 * i]     = qkv_w[row * 192 + col];
    wq[2 * i + 1] = qkv_w[(row + 1) * 192 + col];
  }
  for (int i = tid; i < 4 * 16 * 32; i += 256) {
    int n = i >> 9, k4 = (i >> 5) & 15, ln = i & 31;
    int row = 4 * k4 + 2 * (ln >> 4), col = n * 16 + (ln & 15);
    pw[2 * i]     = proj_w[row * 64 + col];
    pw[2 * i + 1] = proj_w[(row + 1) * 64 + col];
  }
  // bias[h][q][k] = bias_table[rel_idx[h*8+q]][k]  (faithful to reference)
  for (int i = tid; i < 512; i += 256)
    bia[i] = bias_table[rel_idx[i >> 3] * 8 + (i & 7)];
  for (int i = tid; i < 192; i += 256) qkb[i] = qkv_b[i];
  for (int i = tid; i < 64; i += 256) prb[i] = proj_b[i];
  __syncthreads();

  float* xo = wav + wv * WAVE_F;  // 16 x XS : x tile, later attn-out tile
  float* qk = xo + 16 * XS;       // 16 x QS : fused QKV result
  float* pl = qk + 16 * QS;       // 16 x PS : softmax probabilities
  float* mk = pl + 16 * PS;       // 128     : mask rows for the 2 windows

  // zero pad columns 200..207 of qk (zero-feed for N>=8 in P@V B-frags)
  for (int i = lane; i < 128; i += 32) qk[(i >> 3) * QS + 200 + (i & 7)] = 0.f;

  const int tile = blockIdx.x * 8 + wv;   // 65536 tiles total
  const int w0   = tile * 2;              // first window of this tile
  const float* xg = x + (size_t)w0 * 512; // 16 rows x 64 ch, contiguous

  // ---- stage x tile (4 KB) into LDS (async DMA path when available) ----
#ifdef HAVE_ASYNC_LDS
  {
    // builtin wants: (int4 AS1* src, int4 AS3* dst, imm offset, imm cpol)
    typedef v4i __attribute__((address_space(1)))* gas1_t;
    typedef v4i __attribute__((address_space(3)))* las3_t;
#pragma unroll
    for (int j = 0; j < 8; ++j) {
      int i = (j * 32 + lane) * 4;
      __builtin_amdgcn_global_load_async_to_lds_b128(
          (gas1_t)(xg + i), (las3_t)(xo + (i >> 6) * XS + (i & 63)), 0, 0);
    }
  }
#else
#pragma unroll
  for (int j = 0; j < 8; ++j) {
    int i = (j * 32 + lane) * 4;
    *(float4*)(xo + (i >> 6) * XS + (i & 63)) = *(const float4*)(xg + i);
  }
#endif

  // ---- stage mask rows: window n uses mask[n % 32768] ----
  {
    int i = lane * 4;
    int winm = (w0 + (i >> 6)) & 32767;
    *(float4*)(mk + i) = *(const float4*)(mask + (size_t)winm * 64 + (i & 63));
  }

#ifdef HAVE_ASYNC_LDS
# if defined(__has_builtin) && __has_builtin(__builtin_amdgcn_s_wait_asynccnt)
  __builtin_amdgcn_s_wait_asynccnt(0);
# else
  asm volatile("s_wait_asynccnt 0" ::: "memory");
# endif
#endif

  // ---- load x A-fragments (16x64, all 16 K-steps) into registers ----
  v2f ax[16];
#pragma unroll
  for (int k4 = 0; k4 < 16; ++k4)
    ax[k4] = *(const v2f*)(xo + mrow * XS + k4 * 4 + 2 * hi);

  // ---- QKV: (16x64) @ (64x192) + b, 12 N-tiles x 16 K-steps ----
  for (int n = 0; n < 12; ++n) {
    v8f acc = splat8(qkb[n * 16 + mrow]);
#pragma unroll
    for (int k4 = 0; k4 < 16; ++k4)
      acc = wmma4(ax[k4], ((const v2f*)wq)[(n * 16 + k4) * 32 + lane], acc);
#pragma unroll
    for (int jv = 0; jv < 8; ++jv)
      qk[(jv + 8 * hi) * QS + n * 16 + mrow] = acc[jv];
  }

  // ---- attention: per head, 2-window 16x16 tile, block-diagonal valid ----
  const float SCALE = 0.35355339059327373f;           // 8^-0.5
  const bool valid = ((mrow >> 3) == hi);             // same-window element
  const int kk = lane & 7;
  for (int h = 0; h < 8; ++h) {
    // logits = Q @ K^T  (raw; scale/bias/mask folded into epilogue)
    v8f lg = splat8(0.f);
#pragma unroll
    for (int k4 = 0; k4 < 2; ++k4) {
      v2f aq = *(const v2f*)(qk + mrow * QS + h * 8 + k4 * 4 + 2 * hi);
      v2f bk = *(const v2f*)(qk + mrow * QS + 64 + h * 8 + k4 * 4 + 2 * hi);
      lg = wmma4(aq, bk, lg);
    }
    // softmax across the 8 key lanes (xor-shuffle groups of 8, wave32)
#pragma unroll
    for (int jv = 0; jv < 8; ++jv) {
      float l = lg[jv] * SCALE + bia[h * 64 + jv * 8 + kk] +
                mk[hi * 64 + jv * 8 + kk];
      float m = l;
      m = fmaxf(m, __shfl_xor(m, 4));
      m = fmaxf(m, __shfl_xor(m, 2));
      m = fmaxf(m, __shfl_xor(m, 1));
      float e = __expf(l - m);
      float s = e;
      s += __shfl_xor(s, 4);
      s += __shfl_xor(s, 2);
      s += __shfl_xor(s, 1);
      float p = valid ? (e / s) : 0.f;  // zero cross-window block
      pl[(jv + 8 * hi) * PS + mrow] = p;
    }
    // out_h = P (16x16) @ V (16x8, zero-padded to 16 cols)
    v8f ov = splat8(0.f);
    const int cd = (mrow < 8) ? (128 + h * 8 + mrow) : 200;  // 200.. = zeros
#pragma unroll
    for (int k4 = 0; k4 < 4; ++k4) {
      v2f ap = *(const v2f*)(pl + mrow * PS + k4 * 4 + 2 * hi);
      v2f bv;
      bv.x = qk[(k4 * 4 + 2 * hi) * QS + cd];
      bv.y = qk[(k4 * 4 + 2 * hi + 1) * QS + cd];
      ov = wmma4(ap, bv, ov);
    }
    if (mrow < 8) {
#pragma unroll
      for (int jv = 0; jv < 8; ++jv)
        xo[(jv + 8 * hi) * XS + h * 8 + mrow] = ov[jv];  // reuse x buffer
    }
  }

  // ---- proj: (16x64) @ (64x64) + b, streamed straight to global ----
  for (int n = 0; n < 4; ++n) {
    v8f acc = splat8(prb[n * 16 + mrow]);
#pragma unroll
    for (int k4 = 0; k4 < 16; ++k4) {
      v2f ao = *(const v2f*)(xo + mrow * XS + k4 * 4 + 2 * hi);
      acc = wmma4(ao, ((const v2f*)pw)[(n * 16 + k4) * 32 + lane], acc);
    }
#pragma unroll
    for (int jv = 0; jv < 8; ++jv)
      out[(size_t)(w0 * 8 + jv + 8 * hi) * 64 + n * 16 + mrow] = acc[jv];
  }
}

extern "C" void kernel_launch(void* const* d_in, const int* in_sizes, int n_in,
                              void* d_out, int out_size, void* d_ws,
                              size_t ws_size, hipStream_t stream) {
  const float* x          = (const float*)d_in[0];
  const float* mask       = (const float*)d_in[1];
  const float* qkv_w      = (const float*)d_in[2];
  const float* qkv_b      = (const float*)d_in[3];
  const float* proj_w     = (const float*)d_in[4];
  const float* proj_b     = (const float*)d_in[5];
  const float* bias_table = (const float*)d_in[6];
  const int*   rel_idx    = (const int*)d_in[7];
  float* out = (float*)d_out;

  const size_t shmem = (size_t)TOTAL_F * sizeof(float);  // ~220 KB of 320 KB
  (void)hipFuncSetAttribute(reinterpret_cast<const void*>(wattn_kernel),
                            hipFuncAttributeMaxDynamicSharedMemorySize,
                            (int)shmem);
  // 131072 windows / (2 per wave * 8 waves) = 8192 blocks
  wattn_kernel<<<dim3(8192), dim3(256), shmem, stream>>>(
      x, mask, qkv_w, qkv_b, proj_w, proj_b, bias_table, rel_idx, out);
}